// GCN_70763881169543
// MI455X (gfx1250) — compile-verified
//
#include <hip/hip_runtime.h>
#include <hip/hip_bf16.h>

// ---------------------------------------------------------------------------
// GCN (3x GCNConv+BN+LeakyReLU) + mean pool + linear head, fp32, CDNA5.
// Dense parts use V_WMMA_F32_16X16X4_F32 (exact fp32 matrix pipe).
// W tiles staged into LDS with GLOBAL_LOAD_ASYNC_TO_LDS_B128 (ASYNCcnt path).
// ---------------------------------------------------------------------------

typedef float v2f __attribute__((ext_vector_type(2)));
typedef float v8f __attribute__((ext_vector_type(8)));

#define HDIM 128

__device__ __forceinline__ void atomAddF(float* p, float v) {
  unsafeAtomicAdd(p, v);   // lowers to global_atomic_add_f32 on gfx1250
}

// ---------------- degree / normalization precompute ----------------

__global__ void k_deg_init(float* deg, int N) {
  int i = blockIdx.x * blockDim.x + threadIdx.x;
  if (i < N) deg[i] = 1.0f;                       // self-loop
}

__global__ void k_deg_acc(const int* __restrict__ dst, float* deg, int E) {
  int e = blockIdx.x * blockDim.x + threadIdx.x;
  if (e < E) atomAddF(&deg[dst[e]], 1.0f);
}

__global__ void k_dis(const float* __restrict__ deg, float* dis, int N) {
  int i = blockIdx.x * blockDim.x + threadIdx.x;
  if (i < N) dis[i] = rsqrtf(deg[i]);
}

__global__ void k_enorm(const int* __restrict__ src, const int* __restrict__ dst,
                        const float* __restrict__ dis, float* en, int E) {
  int e = blockIdx.x * blockDim.x + threadIdx.x;
  if (e < E) en[e] = dis[src[e]] * dis[dst[e]];
}

// ---------------- WMMA GEMM: out[N,128] = A[N,128] @ W[128,128] -------------
// Block: 256 threads = 8 waves; wave w owns M-tile (blockIdx.x*8+w)*16,
// blockIdx.y owns the 16-wide N tile. W tile (128x16, 8 KB) staged in LDS
// via async-to-LDS loads (ASYNCcnt), no VGPR round-trip.

__global__ void __launch_bounds__(256)
k_gemm128_wmma(const float* __restrict__ A, const float* __restrict__ W,
               float* __restrict__ out, int N) {
  __shared__ float sB[HDIM * 16];                 // sB[k][n], k=0..127, n=0..15
  const int n0 = blockIdx.y << 4;

  // Stage W[k][n0..n0+15] -> sB with GLOBAL_LOAD_ASYNC_TO_LDS_B128.
  // 2048 dwords = 512 x b128 chunks; chunk c: k = c>>2, n = (c&3)*4.
  for (int c = threadIdx.x; c < 512; c += 256) {
    int k = c >> 2, n = (c & 3) << 2;
    const float* gp = W + (k << 7) + n0 + n;      // 16B-aligned
    unsigned lo = (unsigned)(size_t)(&sB[c << 2]);// LDS byte address, 16B-aligned
    asm volatile("global_load_async_to_lds_b128 %0, %1, off"
                 :: "v"(lo), "v"(gp) : "memory");
  }
  asm volatile("s_wait_asynccnt 0x0" ::: "memory");
  __syncthreads();

  const int wave = threadIdx.x >> 5;
  const int lane = threadIdx.x & 31;
  const int lo   = lane & 15;                     // A row-in-tile / B+D column
  const int hi   = lane >> 4;                     // half select (K pairs / D row half)
  const int m0   = (blockIdx.x * 8 + wave) << 4;
  if (m0 >= N) return;                            // wave-uniform exit

  int row = m0 + lo;
  if (row >= N) row = N - 1;                      // clamp loads; stores masked below
  const float* ar = A + (size_t)row * HDIM + 2 * hi;

  v8f acc = {};
#pragma unroll 4
  for (int kk = 0; kk < HDIM; kk += 4) {
    // A 16x4 f32: lane(lo,hi) -> {A[M][kk+2h], A[M][kk+2h+1]}
    v2f a = *(const v2f*)(ar + kk);
    // B 4x16 f32: lane(lo,hi) -> {W[kk+2h][n0+lo], W[kk+2h+1][n0+lo]}
    v2f b;
    b.x = sB[((kk + 2 * hi) << 4) + lo];
    b.y = sB[((kk + 2 * hi + 1) << 4) + lo];
    acc = __builtin_amdgcn_wmma_f32_16x16x4_f32(
        /*neg_a=*/false, a, /*neg_b=*/false, b,
        /*c_mod=*/(short)0, acc, /*reuse_a=*/false, /*reuse_b=*/false);
  }

  // D 16x16 f32: VGPR v -> row m0 + v + 8*hi, col n0 + lo
  const int rb = m0 + 8 * hi;
  float* orow = out + (size_t)rb * HDIM + n0 + lo;
  if (m0 + 16 <= N) {                             // uniform fast path: full tile
#pragma unroll
    for (int v = 0; v < 8; ++v) orow[(size_t)v * HDIM] = acc[v];
  } else {                                        // only the last M-tile
#pragma unroll
    for (int v = 0; v < 8; ++v)
      if (rb + v < N) orow[(size_t)v * HDIM] = acc[v];
  }
}

// ---------------- per-layer aggregation ----------------

// agg[i] = hlin[i] * self_norm(node) + bias(feature)   (full overwrite)
__global__ void k_agg_init(const float* __restrict__ lin, const float* __restrict__ dis,
                           const float* __restrict__ b, float* agg, int N) {
  size_t i = (size_t)blockIdx.x * blockDim.x + threadIdx.x;
  if (i < (size_t)N * HDIM) {
    int n = (int)(i >> 7), f = (int)(i & 127);
    float d = dis[n];
    agg[i] = lin[i] * d * d + b[f];
  }
}

// 32 lanes per edge, 4 features per lane: gather h[src], scale, atomic scatter.
__global__ void k_scatter(const float* __restrict__ hlin, const int* __restrict__ src,
                          const int* __restrict__ dst, const float* __restrict__ en,
                          float* agg, int E) {
  long long gid = (long long)blockIdx.x * blockDim.x + threadIdx.x;
  long long e = gid >> 5;
  if (e >= E) return;
  int f = ((int)gid & 31) << 2;
  float w = en[e];
  int s = src[e], d = dst[e];
  float4 hv = *(const float4*)(hlin + (size_t)s * HDIM + f);
  float* o = agg + (size_t)d * HDIM + f;
  atomAddF(o + 0, hv.x * w);
  atomAddF(o + 1, hv.y * w);
  atomAddF(o + 2, hv.z * w);
  atomAddF(o + 3, hv.w * w);
}

// ---------------- batch norm ----------------

__global__ void k_zero(float* a, int n) {
  int i = blockIdx.x * blockDim.x + threadIdx.x;
  if (i < n) a[i] = 0.0f;
}

__global__ void k_bn_stats(const float* __restrict__ agg, float* sums, float* sumsq, int N) {
  int f = threadIdx.x;                            // 128 threads = one feature each
  float s = 0.0f, q = 0.0f;
  for (int r = blockIdx.x; r < N; r += gridDim.x) {
    float v = agg[(size_t)r * HDIM + f];
    s += v;
    q += v * v;
  }
  atomAddF(&sums[f], s);
  atomAddF(&sumsq[f], q);
}

__global__ void k_bn_fin(const float* __restrict__ sums, const float* __restrict__ sumsq,
                         const float* __restrict__ g, const float* __restrict__ be,
                         float* scale, float* shift, int N) {
  int f = threadIdx.x;
  if (f < HDIM) {
    float inv  = 1.0f / (float)N;
    float mean = sums[f] * inv;
    float var  = sumsq[f] * inv - mean * mean;    // biased, matches reference
    float sc   = rsqrtf(var + 1e-5f) * g[f];
    scale[f] = sc;
    shift[f] = be[f] - mean * sc;
  }
}

// in-place BN affine + leaky_relu(0.1)
__global__ void k_bn_apply(float* agg, const float* __restrict__ scale,
                           const float* __restrict__ shift, int N) {
  size_t i = (size_t)blockIdx.x * blockDim.x + threadIdx.x;
  if (i < (size_t)N * HDIM) {
    int f = (int)(i & 127);
    float v = agg[i] * scale[f] + shift[f];
    agg[i] = v > 0.0f ? v : 0.1f * v;
  }
}

// ---------------- pooling + head ----------------

__global__ void k_pool(const float* __restrict__ h, const int* __restrict__ batch,
                       float* pooled, float* cnt, int N) {
  size_t i = (size_t)blockIdx.x * blockDim.x + threadIdx.x;
  if (i < (size_t)N * HDIM) {
    int n = (int)(i >> 7), f = (int)(i & 127);
    int b = batch[n];
    atomAddF(&pooled[(size_t)b * HDIM + f], h[i]);
    if (f == 0) atomAddF(&cnt[b], 1.0f);
  }
}

__global__ void k_final(const float* __restrict__ pooled, const float* __restrict__ cnt,
                        const float* __restrict__ Wl, const float* __restrict__ bl,
                        float* out, int C) {
  int g = blockIdx.x;
  int c = threadIdx.x;
  if (c >= C) return;
  float inv = 1.0f / fmaxf(cnt[g], 1.0f);
  float acc = bl[c];
#pragma unroll 8
  for (int k = 0; k < HDIM; ++k)
    acc += pooled[(size_t)g * HDIM + k] * inv * Wl[k * C + c];
  out[g * C + c] = acc;
}

// ---------------------------------------------------------------------------

static inline int cdiv(long long a, long long b) { return (int)((a + b - 1) / b); }

extern "C" void kernel_launch(void* const* d_in, const int* in_sizes, int n_in,
                              void* d_out, int out_size, void* d_ws, size_t ws_size,
                              hipStream_t stream) {
  const float* x     = (const float*)d_in[0];
  const int*   ei    = (const int*)d_in[1];
  const int*   batch = (const int*)d_in[2];
  const float* Wv[3] = {(const float*)d_in[3], (const float*)d_in[7], (const float*)d_in[11]};
  const float* bv[3] = {(const float*)d_in[4], (const float*)d_in[8], (const float*)d_in[12]};
  const float* gv[3] = {(const float*)d_in[5], (const float*)d_in[9], (const float*)d_in[13]};
  const float* ev[3] = {(const float*)d_in[6], (const float*)d_in[10], (const float*)d_in[14]};
  const float* Wl = (const float*)d_in[15];
  const float* bl = (const float*)d_in[16];
  float* out = (float*)d_out;

  const int N = in_sizes[0] / HDIM;
  const int E = in_sizes[1] / 2;
  const int C = in_sizes[16];
  const int G = out_size / C;                     // NUM_GRAPHS
  const int* src = ei;
  const int* dst = ei + E;

  // workspace carve (floats)
  float* ws     = (float*)d_ws;
  float* bufB   = ws;                             // N*128  (h = in @ W)
  float* bufC   = bufB + (size_t)N * HDIM;        // N*128  (agg / layer output)
  float* deg    = bufC + (size_t)N * HDIM;        // N
  float* dis    = deg + N;                        // N
  float* enorm  = dis + N;                        // E
  float* sums   = enorm + E;                      // 128
  float* sumsq  = sums + HDIM;                    // 128
  float* scale  = sumsq + HDIM;                   // 128
  float* shift  = scale + HDIM;                   // 128
  float* pooled = shift + HDIM;                   // G*128
  float* cnt    = pooled + (size_t)G * HDIM;      // G

  const size_t NF = (size_t)N * HDIM;
  const int TB = 256;

  // --- degree / norms (shared by all layers) ---
  k_deg_init<<<cdiv(N, TB), TB, 0, stream>>>(deg, N);
  k_deg_acc<<<cdiv(E, TB), TB, 0, stream>>>(dst, deg, E);
  k_dis<<<cdiv(N, TB), TB, 0, stream>>>(deg, dis, N);
  k_enorm<<<cdiv(E, TB), TB, 0, stream>>>(src, dst, dis, enorm, E);

  // --- three GCN blocks: in -> bufB (gemm) -> bufC (agg, BN in place) ---
  const float* in = x;
  for (int l = 0; l < 3; ++l) {
    dim3 gg(cdiv(N, 128), HDIM / 16);
    k_gemm128_wmma<<<gg, TB, 0, stream>>>(in, Wv[l], bufB, N);
    k_agg_init<<<cdiv((long long)NF, TB), TB, 0, stream>>>(bufB, dis, bv[l], bufC, N);
    k_scatter<<<cdiv((long long)E * 32, TB), TB, 0, stream>>>(bufB, src, dst, enorm, bufC, E);
    k_zero<<<1, 256, 0, stream>>>(sums, 2 * HDIM);            // sums+sumsq contiguous
    k_bn_stats<<<512, HDIM, 0, stream>>>(bufC, sums, sumsq, N);
    k_bn_fin<<<1, HDIM, 0, stream>>>(sums, sumsq, gv[l], ev[l], scale, shift, N);
    k_bn_apply<<<cdiv((long long)NF, TB), TB, 0, stream>>>(bufC, scale, shift, N);
    in = bufC;
  }

  // --- global mean pool + linear head ---
  k_zero<<<cdiv(G * HDIM + G, TB), TB, 0, stream>>>(pooled, G * HDIM + G);
  k_pool<<<cdiv((long long)NF, TB), TB, 0, stream>>>(bufC, batch, pooled, cnt, N);
  k_final<<<G, 32, 0, stream>>>(pooled, cnt, Wl, bl, out, C);
}